// PPRPowerIteration_24257975288205
// MI455X (gfx1250) — compile-verified
//
#include <hip/hip_runtime.h>
#include <cstdint>

#define FEAT   128
#define FEAT4  32          // FEAT / 4
#define ALPHA_C 0.1f
#define NITER_C 10

// ---------------------------------------------------------------------------
// CSR build: zero counts -> histogram -> single-block scan -> scatter
// ---------------------------------------------------------------------------
__global__ __launch_bounds__(256) void zero_i32_kernel(int* __restrict__ p, int n) {
  int i = blockIdx.x * 256 + threadIdx.x;
  if (i < n) p[i] = 0;
}

__global__ __launch_bounds__(256) void hist_kernel(const int* __restrict__ rows,
                                                   int* __restrict__ counts, int E) {
  int e = blockIdx.x * 256 + threadIdx.x;
  if (e < E) atomicAdd(&counts[rows[e]], 1);
}

// Exclusive scan of cnt[0..n) -> offs[0..n]; also resets cnt[i] = offs[i] (cursor).
__global__ __launch_bounds__(1024) void scan_kernel(int* __restrict__ cnt,
                                                    int* __restrict__ offs, int n) {
  __shared__ int sm[1024];
  __shared__ int carry_s;
  if (threadIdx.x == 0) carry_s = 0;
  __syncthreads();
  for (int base = 0; base < n; base += 1024) {
    int i = base + (int)threadIdx.x;
    int v = (i < n) ? cnt[i] : 0;
    sm[threadIdx.x] = v;
    __syncthreads();
    for (int o = 1; o < 1024; o <<= 1) {
      int t = (threadIdx.x >= (unsigned)o) ? sm[threadIdx.x - o] : 0;
      __syncthreads();
      sm[threadIdx.x] += t;
      __syncthreads();
    }
    int carry = carry_s;
    int excl  = carry + sm[threadIdx.x] - v;
    if (i < n) { offs[i] = excl; cnt[i] = excl; }
    __syncthreads();
    if (threadIdx.x == 1023) carry_s = carry + sm[1023];
    __syncthreads();
  }
  if (threadIdx.x == 0) offs[n] = carry_s;
}

// Pack each edge as {val_bits[63:32] | col[31:0]} into its row's CSR segment.
__global__ __launch_bounds__(256) void scatter_kernel(const int* __restrict__ rows,
                                                      const int* __restrict__ cols,
                                                      const float* __restrict__ vals,
                                                      int* __restrict__ cursor,
                                                      unsigned long long* __restrict__ csr,
                                                      int E) {
  int e = blockIdx.x * 256 + threadIdx.x;
  if (e < E) {
    int r   = rows[e];
    int pos = atomicAdd(&cursor[r], 1);
    unsigned long long packed =
        ((unsigned long long)__float_as_uint(vals[e]) << 32) | (unsigned)cols[e];
    csr[pos] = packed;
  }
}

// ---------------------------------------------------------------------------
// SpMM: one wave32 per row, lane owns 4 features (float4).
// Edge list staged into LDS via double-buffered CDNA5 async global->LDS copies.
// nxt[row,:] = alpha*local[row,:] + sum_e val_e * cur[col_e,:]
// ---------------------------------------------------------------------------
__global__ __launch_bounds__(256) void spmm_csr_kernel(
    const unsigned long long* __restrict__ csr,
    const int* __restrict__ offs,
    const float4* __restrict__ cur4,
    const float4* __restrict__ loc4,
    float4* __restrict__ nxt4,
    int n) {
  __shared__ unsigned long long stage[8][2][32];   // per-wave 2 x 256B edge stage
  const int wave = threadIdx.x >> 5;
  const int lane = threadIdx.x & 31;
  const int row  = blockIdx.x * 8 + wave;
  if (row >= n) return;

  const int beg = offs[row];
  const int end = offs[row + 1];

  // LDS byte offsets of this lane's two staging slots
  // (generic-pointer low 32 bits == DS byte offset per the LDS aperture rule)
  const unsigned lds0 = (unsigned)(size_t)&stage[wave][0][lane];
  const unsigned lds1 = (unsigned)(size_t)&stage[wave][1][lane];

  // Kick off the first chunk's DMA before the accumulator-init load so the
  // async copy overlaps it.
  if (beg < end) {
    int e = beg + lane;
    if (e < end) {
      const unsigned long long* gp = csr + e;
      asm volatile("global_load_async_to_lds_b64 %0, %1, off"
                   :: "v"(lds0), "v"(gp) : "memory");
    }
  }

  float4 l = loc4[(size_t)row * FEAT4 + lane];
  float4 acc = make_float4(ALPHA_C * l.x, ALPHA_C * l.y, ALPHA_C * l.z, ALPHA_C * l.w);

  int buf = 0;
  for (int c = beg; c < end; c += 32) {
    const int nc = c + 32;
    if (nc < end) {
      // Prefetch next chunk into the other buffer, then wait only for the
      // older copy (async loads complete in order -> asynccnt<=1 suffices).
      int e = nc + lane;
      unsigned dst = buf ? lds0 : lds1;
      if (e < end) {
        const unsigned long long* gp = csr + e;
        asm volatile("global_load_async_to_lds_b64 %0, %1, off"
                     :: "v"(dst), "v"(gp) : "memory");
      }
      asm volatile("s_wait_asynccnt 0x1" ::: "memory");
    } else {
      asm volatile("s_wait_asynccnt 0x0" ::: "memory");
    }

    int cnt = end - c; if (cnt > 32) cnt = 32;
#pragma unroll 8
    for (int j = 0; j < cnt; ++j) {
      unsigned long long p = stage[wave][buf][j];
      int   col = (int)(unsigned)(p & 0xffffffffULL);
      float v   = __uint_as_float((unsigned)(p >> 32));
      float4 x  = cur4[(size_t)col * FEAT4 + lane];
      acc.x = fmaf(v, x.x, acc.x);
      acc.y = fmaf(v, x.y, acc.y);
      acc.z = fmaf(v, x.z, acc.z);
      acc.w = fmaf(v, x.w, acc.w);
    }
    buf ^= 1;
  }
  nxt4[(size_t)row * FEAT4 + lane] = acc;
}

// ---------------------------------------------------------------------------
// Fallback (workspace too small for CSR): atomic edge scatter
// ---------------------------------------------------------------------------
__global__ __launch_bounds__(256) void init_alpha_kernel(const float4* __restrict__ loc4,
                                                         float4* __restrict__ nxt4, int n4) {
  int i = blockIdx.x * 256 + threadIdx.x;
  if (i < n4) {
    float4 l = loc4[i];
    nxt4[i] = make_float4(ALPHA_C * l.x, ALPHA_C * l.y, ALPHA_C * l.z, ALPHA_C * l.w);
  }
}

__global__ __launch_bounds__(256) void edge_atomic_kernel(const int* __restrict__ rows,
                                                          const int* __restrict__ cols,
                                                          const float* __restrict__ vals,
                                                          const float4* __restrict__ cur4,
                                                          float* __restrict__ nxt, int E) {
  const int wave = threadIdx.x >> 5;
  const int lane = threadIdx.x & 31;
  int e = blockIdx.x * 8 + wave;
  if (e >= E) return;
  int   r   = rows[e];
  int   col = cols[e];
  float v   = vals[e];
  float4 x  = cur4[(size_t)col * FEAT4 + lane];
  float* dst = nxt + (size_t)r * FEAT + lane * 4;
  atomicAdd(dst + 0, v * x.x);
  atomicAdd(dst + 1, v * x.y);
  atomicAdd(dst + 2, v * x.z);
  atomicAdd(dst + 3, v * x.w);
}

// ---------------------------------------------------------------------------
// Final gather: out[i,:] = preds[idx[i],:]
// ---------------------------------------------------------------------------
__global__ __launch_bounds__(256) void gather_kernel(const float4* __restrict__ src4,
                                                     const int* __restrict__ idx,
                                                     float4* __restrict__ out4, int m) {
  const int wave = threadIdx.x >> 5;
  const int lane = threadIdx.x & 31;
  int i = blockIdx.x * 8 + wave;
  if (i >= m) return;
  out4[(size_t)i * FEAT4 + lane] = src4[(size_t)idx[i] * FEAT4 + lane];
}

// ---------------------------------------------------------------------------
extern "C" void kernel_launch(void* const* d_in, const int* in_sizes, int n_in,
                              void* d_out, int out_size, void* d_ws, size_t ws_size,
                              hipStream_t stream) {
  const float* local = (const float*)d_in[0];
  const float* vals  = (const float*)d_in[1];
  const int*   rows  = (const int*)d_in[2];
  const int*   cols  = (const int*)d_in[3];
  const int*   idx   = (const int*)d_in[4];

  const int E = in_sizes[1];
  const int N = in_sizes[0] / FEAT;
  const int M = in_sizes[4];

  const size_t buf_elems = (size_t)N * FEAT;
  char* ws = (char*)d_ws;

  // Layout: [csr u64 * E][bufA f32 * N*F][bufB f32 * N*F][offs i32 * (N+1)][cursor i32 * N]
  const size_t need_csr = (size_t)E * 8 + buf_elems * 8 + ((size_t)2 * N + 1) * 4;
  const bool use_csr = (ws_size >= need_csr);

  dim3 blk(256);

  if (use_csr) {
    unsigned long long* csr = (unsigned long long*)ws;
    float* bufA = (float*)(ws + (size_t)E * 8);
    float* bufB = bufA + buf_elems;
    int*   offs = (int*)(bufB + buf_elems);
    int*   curs = offs + (N + 1);

    zero_i32_kernel<<<(N + 255) / 256, blk, 0, stream>>>(curs, N);
    hist_kernel<<<(E + 255) / 256, blk, 0, stream>>>(rows, curs, E);
    scan_kernel<<<1, 1024, 0, stream>>>(curs, offs, N);
    scatter_kernel<<<(E + 255) / 256, blk, 0, stream>>>(rows, cols, vals, curs, csr, E);

    const float* cur = local;
    float*       nxt = bufA;
    for (int t = 0; t < NITER_C; ++t) {
      spmm_csr_kernel<<<(N + 7) / 8, blk, 0, stream>>>(
          csr, offs, (const float4*)cur, (const float4*)local, (float4*)nxt, N);
      cur = nxt;
      nxt = (cur == bufA) ? bufB : bufA;
    }
    gather_kernel<<<(M + 7) / 8, blk, 0, stream>>>(
        (const float4*)cur, idx, (float4*)d_out, M);
  } else {
    // Atomic fallback: only needs the two ping-pong buffers.
    float* bufA = (float*)ws;
    float* bufB = bufA + buf_elems;
    const int n4 = (int)(buf_elems / 4);

    const float* cur = local;
    float*       nxt = bufA;
    for (int t = 0; t < NITER_C; ++t) {
      init_alpha_kernel<<<(n4 + 255) / 256, blk, 0, stream>>>(
          (const float4*)local, (float4*)nxt, n4);
      edge_atomic_kernel<<<(E + 7) / 8, blk, 0, stream>>>(
          rows, cols, vals, (const float4*)cur, nxt, E);
      cur = nxt;
      nxt = (cur == bufA) ? bufB : bufA;
    }
    gather_kernel<<<(M + 7) / 8, blk, 0, stream>>>(
        (const float4*)cur, idx, (float4*)d_out, M);
  }
}